// MarchingTetrahedraHelper_7962869367257
// MI455X (gfx1250) — compile-verified
//
#include <hip/hip_runtime.h>
#include <hip/hip_bf16.h>
#include <stdint.h>

typedef unsigned int       u32;
typedef unsigned long long u64;
typedef unsigned char      u8;

typedef unsigned int u32x4 __attribute__((ext_vector_type(4)));
typedef int          i32x4 __attribute__((ext_vector_type(4)));
typedef int          i32x8 __attribute__((ext_vector_type(8)));

// ---------------- marching-tets tables ----------------
__constant__ int c_TRI[16][6] = {
 {-1,-1,-1,-1,-1,-1},{1,0,2,-1,-1,-1},{4,0,3,-1,-1,-1},{1,4,2,1,3,4},
 {3,1,5,-1,-1,-1},{2,3,0,2,5,3},{1,4,0,1,5,4},{4,2,5,-1,-1,-1},
 {4,5,2,-1,-1,-1},{4,1,0,4,5,1},{3,2,0,3,5,2},{1,3,5,-1,-1,-1},
 {4,1,2,4,3,1},{3,0,4,-1,-1,-1},{2,0,1,-1,-1,-1},{-1,-1,-1,-1,-1,-1}};
__constant__ int c_NUM[16]  = {0,1,1,2,1,2,2,1,1,2,2,1,2,1,1,0};
__constant__ int c_EDGE[12] = {0,1,0,2,0,3,1,2,1,3,2,3};

// =====================================================================
// Kernel 1: per-vertex crossing mask (7 monotone directions, sorted by
// linear offset => lexicographic edge order) + crossing count.
// =====================================================================
__global__ __launch_bounds__(256)
void k_vertex(const float* __restrict__ level, const int* __restrict__ thrp,
              u32* __restrict__ vmask, u32* __restrict__ vcnt, int n, int N) {
  const int v = blockIdx.x * 256 + threadIdx.x;
  if (v >= N) return;
  const float thr = (float)thrp[0];
  const int n2 = n * n;
  const int k = v % n, j = (v / n) % n, i = v / n2;
  // CDNA5: global_prefetch_b8 on the streamed level array (one stride ahead)
  {
    int pf = v + n2 + n + 1 + 2048; if (pf >= N) pf = N - 1;
    __builtin_prefetch(level + pf, 0, 1);
  }
  const float l0 = level[v];
  const bool  o0 = l0 > thr;
  const int offs[7] = {1, n, n + 1, n2, n2 + 1, n2 + n, n2 + n + 1};
  u32 m = 0;
#pragma unroll
  for (int d = 0; d < 7; ++d) {
    const int c  = d + 1;                       // (di,dj,dk) bits
    const int di = c >> 2, dj = (c >> 1) & 1, dk = c & 1;
    if ((i + di < n) && (j + dj < n) && (k + dk < n)) {
      const bool o1 = level[v + offs[d]] > thr;
      if (o1 != o0) m |= (1u << d);
    }
  }
  vmask[v] = m;
  vcnt[v]  = (u32)__popc(m);
}

// =====================================================================
// Hierarchical exclusive scan (deterministic, no atomics).
// NOTE: in/out may alias (called in-place) -> no __restrict__ on them.
// =====================================================================
template <typename T>
__global__ __launch_bounds__(256)
void k_scan_block(const T* in, T* out, T* __restrict__ blockSums, int nElems) {
  __shared__ T sm[256];
  const int tid = threadIdx.x;
  const int g   = blockIdx.x * 256 + tid;
  const T v = (g < nElems) ? in[g] : (T)0;
  sm[tid] = v;
  __syncthreads();
#pragma unroll
  for (int off = 1; off < 256; off <<= 1) {
    const T add = (tid >= off) ? sm[tid - off] : (T)0;
    __syncthreads();
    sm[tid] += add;
    __syncthreads();
  }
  const T incl = sm[tid];
  if (g < nElems) out[g] = incl - v;            // exclusive
  if (tid == 255) blockSums[blockIdx.x] = incl; // block total
}

template <typename T>
__global__ __launch_bounds__(256)
void k_scan_add(T* __restrict__ data, const T* __restrict__ offs, int nElems) {
  const int g = blockIdx.x * 256 + threadIdx.x;
  if (g < nElems) data[g] += offs[blockIdx.x];
}

// =====================================================================
// Kernel 3: emit interpolated vertices in global crossing-edge order.
// Output is write-once streaming data -> non-temporal stores (TH=NT).
// =====================================================================
__global__ __launch_bounds__(256)
void k_emit_verts(const float* __restrict__ level, const float* __restrict__ pos,
                  const int* __restrict__ thrp, const u32* __restrict__ vmask,
                  const u32* __restrict__ vbase, float* __restrict__ out,
                  int n, int N) {
  const int v = blockIdx.x * 256 + threadIdx.x;
  if (v >= N) return;
  const u32 m = vmask[v];
  if (!m) return;
  const float thr = (float)thrp[0];
  const int n2 = n * n;
  const int offs[7] = {1, n, n + 1, n2, n2 + 1, n2 + n, n2 + n + 1};
  const float s0  = level[v] - thr;
  const float p0x = pos[3 * v], p0y = pos[3 * v + 1], p0z = pos[3 * v + 2];
  u32 r = vbase[v];
#pragma unroll
  for (int d = 0; d < 7; ++d) {
    if (m & (1u << d)) {
      const int u = v + offs[d];
      const float s1 = level[u] - thr;
      const float denom = s0 - s1;        // == s.sum(1) in the reference
      const float w0 = (-s1) / denom;     // s[:, ::-1] / denom
      const float w1 = s0 / denom;
      __builtin_nontemporal_store(p0x * w0 + pos[3 * u + 0] * w1, &out[3 * r + 0]);
      __builtin_nontemporal_store(p0y * w0 + pos[3 * u + 1] * w1, &out[3 * r + 1]);
      __builtin_nontemporal_store(p0z * w0 + pos[3 * u + 2] * w1, &out[3 * r + 2]);
      ++r;
    }
  }
}

// =====================================================================
// TDM helper: 1-D tile (up to 1024 dwords) global -> LDS offset 0.
// D# built per CDNA5 ISA. Issued once per
// wave (EXEC ignored by TDM), waited with s_wait_tensorcnt.
// =====================================================================
__device__ __forceinline__ void tdm_load_tile_1d(const void* gsrc, u32 remainElems) {
#if defined(__HIP_DEVICE_COMPILE__) && defined(__gfx1250__)
  const u64 ga = (u64)gsrc;
  u32x4 g0;
  g0.x = 1u;                                        // count=1 valid descriptor
  g0.y = 0u;                                        // lds_addr = 0 (sole LDS alloc)
  g0.z = (u32)(ga & 0xffffffffu);                   // global_addr[31:0]
  g0.w = (u32)((ga >> 32) & 0x01ffffffu) | (2u << 30); // addr[56:32] | type=2
  const u32 td0 = remainElems;                      // OOB reads return 0
  i32x8 g1;
  g1[0] = (int)(2u << 16);                          // data_size = 4 bytes
  g1[1] = (int)((td0 & 0xffffu) << 16);             // tensor_dim0[15:0]
  g1[2] = (int)(((td0 >> 16) & 0xffffu) | (1u << 16)); // dim0[31:16] | tensor_dim1=1
  g1[3] = (int)(1024u << 16);                       // tile_dim0 = 1024 dwords
  g1[4] = 1;                                        // tile_dim1 = 1
  g1[5] = (int)td0;                                 // tensor_dim0_stride[31:0]
  g1[6] = 0;
  g1[7] = 0;
  const i32x4 z4 = {0, 0, 0, 0};
#if __clang_major__ >= 23
  const i32x8 z8 = {0, 0, 0, 0, 0, 0, 0, 0};
  __builtin_amdgcn_tensor_load_to_lds(g0, g1, z4, z4, z8, 0);
#else
  __builtin_amdgcn_tensor_load_to_lds(g0, g1, z4, z4, 0);
#endif
  __builtin_amdgcn_s_wait_tensorcnt(0);
#else
  (void)gsrc; (void)remainElems;
#endif
}

// =====================================================================
// Kernel 4: classify tets (TDM-staged tet indices), write packed
// {m1 | m2<<32} for the face-slot scan + tetindex byte.
// =====================================================================
__global__ __launch_bounds__(256)
void k_tet_class(const int* __restrict__ tet, const float* __restrict__ level,
                 const int* __restrict__ thrp, u64* __restrict__ tsum,
                 u8* __restrict__ tidx, int F) {
  __shared__ int stet[1024];
  const int t = blockIdx.x * 256 + threadIdx.x;
#if defined(__HIP_DEVICE_COMPILE__) && defined(__gfx1250__)
  if (threadIdx.x < 32) {                 // one TDM issue per block (wave 0)
    const u32 blockStart = (u32)blockIdx.x * 1024u;
    tdm_load_tile_1d(tet + blockStart, 4u * (u32)F - blockStart);
  }
  __syncthreads();
#else
  if (t < F) {
    const int4 q = ((const int4*)tet)[t];
    stet[threadIdx.x * 4 + 0] = q.x; stet[threadIdx.x * 4 + 1] = q.y;
    stet[threadIdx.x * 4 + 2] = q.z; stet[threadIdx.x * 4 + 3] = q.w;
  }
  __syncthreads();
#endif
  if (t >= F) return;
  const float thr = (float)thrp[0];
  const int lt = threadIdx.x * 4;
  const int a = stet[lt], b = stet[lt + 1], c = stet[lt + 2], d = stet[lt + 3];
  const int ti = (level[a] > thr ? 1 : 0) | (level[b] > thr ? 2 : 0) |
                 (level[c] > thr ? 4 : 0) | (level[d] > thr ? 8 : 0);
  const int nt = c_NUM[ti];
  tsum[t] = (nt == 1 ? 1ull : 0ull) | (nt == 2 ? (1ull << 32) : 0ull);
  tidx[t] = (u8)ti;
}

// =====================================================================
// Kernel 6: emit faces. Tet indices staged with per-lane
// global_load_async_to_lds_b128 (ASYNCcnt path). Edge -> global crossing
// rank via vbase[v0] + popc(vmask[v0] & below(dir)). NT stores for output.
// =====================================================================
__global__ __launch_bounds__(256)
void k_emit_faces(const int* __restrict__ tet, const u8* __restrict__ tidx,
                  const u64* __restrict__ tbase, const u32* __restrict__ vbase,
                  const u32* __restrict__ vmask, const u32* __restrict__ Mtot,
                  const u64* __restrict__ Ttot, float* __restrict__ out,
                  int n, int F) {
  __shared__ int stet[1024];
  const int t = blockIdx.x * 256 + threadIdx.x;
#if defined(__HIP_DEVICE_COMPILE__) && defined(__gfx1250__)
  {
    const int ic  = (t < F) ? t : (F - 1);
    const u32 voff = (u32)ic * 16u;           // byte offset into tet[]
    const u32 lds  = (u32)threadIdx.x * 16u;  // LDS byte addr (alloc at 0)
    asm volatile("global_load_async_to_lds_b128 %0, %1, %2"
                 :: "v"(lds), "v"(voff), "s"((u64)tet) : "memory");
    asm volatile("s_wait_asynccnt 0" ::: "memory");
  }
  __syncthreads();
#else
  if (t < F) {
    const int4 q = ((const int4*)tet)[t];
    stet[threadIdx.x * 4 + 0] = q.x; stet[threadIdx.x * 4 + 1] = q.y;
    stet[threadIdx.x * 4 + 2] = q.z; stet[threadIdx.x * 4 + 3] = q.w;
  }
  __syncthreads();
#endif
  if (t >= F) return;
  const int ti = (int)tidx[t];
  const int nt = c_NUM[ti];
  if (!nt) return;
  const u64 bb = tbase[t];
  const u32 b1 = (u32)bb, b2 = (u32)(bb >> 32);
  const u32 S1 = (u32)(Ttot[0] & 0xffffffffu);  // total 1-tri tets
  const u32 faceBase = 3u * Mtot[0];            // faces start after verts
  const int lt = threadIdx.x * 4;
  const int vv[4] = {stet[lt], stet[lt + 1], stet[lt + 2], stet[lt + 3]};
  const int n2 = n * n;
  const u32 fi0 = (nt == 1) ? b1 : (S1 + 2u * b2);
  const int ne = nt * 3;
  for (int e = 0; e < ne; ++e) {
    const int eid = c_TRI[ti][e];
    const int va = vv[c_EDGE[2 * eid]], vb = vv[c_EDGE[2 * eid + 1]];
    const int v0 = va < vb ? va : vb;
    int delta = (va < vb ? vb : va) - v0;
    const int di = (delta >= n2) ? 1 : 0; delta -= di * n2;
    const int dj = (delta >= n)  ? 1 : 0; delta -= dj * n;
    const int d  = ((di << 2) | (dj << 1) | delta) - 1;   // sorted-offset dir id
    const u32 rank = vbase[v0] + (u32)__popc(vmask[v0] & ((1u << d) - 1u));
    const u32 fidx = fi0 + (u32)(e / 3);
    __builtin_nontemporal_store((float)rank,
                                &out[faceBase + 3u * fidx + (u32)(e % 3)]);
  }
}

// =====================================================================
// Host side
// =====================================================================
template <typename T>
static void run_scan(T* data, T* s0, T* s1, T* total, int nElems, hipStream_t stream) {
  const int nb0 = (nElems + 255) / 256;
  const int nb1 = (nb0 + 255) / 256;    // <= 256 for our sizes
  k_scan_block<T><<<nb0, 256, 0, stream>>>(data, data, s0, nElems);
  k_scan_block<T><<<nb1, 256, 0, stream>>>(s0, s0, s1, nb0);
  k_scan_block<T><<<1,   256, 0, stream>>>(s1, s1, total, nb1);
  k_scan_add<T><<<nb1, 256, 0, stream>>>(s0, s1, nb0);
  k_scan_add<T><<<nb0, 256, 0, stream>>>(data, s0, nElems);
}

extern "C" void kernel_launch(void* const* d_in, const int* in_sizes, int n_in,
                              void* d_out, int out_size, void* d_ws, size_t ws_size,
                              hipStream_t stream) {
  const float* level = (const float*)d_in[0];          // [N]
  const float* pos   = (const float*)d_in[1];          // [N,3]
  const int*   tet   = (const int*)d_in[2];            // [F,4] (jax default int32)
  const int*   thrp  = (const int*)d_in[3];            // scalar threshold
  float* outF = (float*)d_out;

  const int N = in_sizes[0];
  const int F = in_sizes[2] / 4;
  int n = 1; while ((long long)n * n * n < (long long)N) ++n;  // (res+1)

  const int nbN  = (N + 255) / 256;
  const int nbF  = (F + 255) / 256;
  const int nb0v = nbN, nb0t = nbF;

  // Workspace carve-out (256B aligned slices)
  char* w = (char*)d_ws;
  auto carve = [&](size_t bytes) -> void* {
    void* p = (void*)w; w += (bytes + 255) & ~(size_t)255; return p;
  };
  u32* vbase = (u32*)carve((size_t)N * 4);    // counts -> exclusive scan
  u32* vmask = (u32*)carve((size_t)N * 4);
  u64* tbase = (u64*)carve((size_t)F * 8);    // packed {m1, m2<<32} -> scan
  u8*  tidx  = (u8*) carve((size_t)F);
  u32* s0v   = (u32*)carve((size_t)nb0v * 4);
  u32* s1v   = (u32*)carve(256 * 4);
  u64* s0t   = (u64*)carve((size_t)nb0t * 8);
  u64* s1t   = (u64*)carve(256 * 8);
  u32* Mtot  = (u32*)carve(256);
  u64* Ttot  = (u64*)carve(256);
  (void)ws_size; (void)n_in; (void)out_size;

  // 1) vertex crossing masks/counts
  k_vertex<<<nbN, 256, 0, stream>>>(level, thrp, vmask, vbase, n, N);
  // 2) vertex rank scan (vbase -> exclusive prefix, Mtot = #crossing edges)
  run_scan<u32>(vbase, s0v, s1v, Mtot, N, stream);
  // 3) interpolated surface vertices
  k_emit_verts<<<nbN, 256, 0, stream>>>(level, pos, thrp, vmask, vbase, outF, n, N);
  // 4) tet classification (TDM-staged)
  k_tet_class<<<nbF, 256, 0, stream>>>(tet, level, thrp, tbase, tidx, F);
  // 5) face-slot scan (Ttot = {S1, S2<<32})
  run_scan<u64>(tbase, s0t, s1t, Ttot, F, stream);
  // 6) faces (async-to-LDS staged)
  k_emit_faces<<<nbF, 256, 0, stream>>>(tet, tidx, tbase, vbase, vmask,
                                        Mtot, Ttot, outF, n, F);
}